// TensorProductAggLayer_20607253086902
// MI455X (gfx1250) — compile-verified
//
#include <hip/hip_runtime.h>
#include <hip/hip_bf16.h>

typedef float v2f __attribute__((ext_vector_type(2)));
typedef float v8f __attribute__((ext_vector_type(8)));
typedef unsigned int u32x4 __attribute__((ext_vector_type(4)));
typedef int i32x4 __attribute__((ext_vector_type(4)));
typedef int i32x8 __attribute__((ext_vector_type(8)));

#define N_NODES 10000
#define N_EDGES 160000
#define TE 64            // edges per workgroup (160000/64 = 2500 blocks, no tail)
#define LDA 132          // padded LDS row stride in floats (128+4 -> conflict-free)

// constants from reference
#define PW0   0.17677669529663688f    /* sqrt(1/32) */
#define PW0I3 0.10206207261596575f    /* PW0 / sqrt(3) */
#define PW1I3 0.17677669529663688f    /* sqrt(3/32)/sqrt(3) == sqrt(1/32) */

// workspace layout (floats)
#define WS_SUM 0
#define WS_CNT (N_NODES * 64)                     // 640000
#define WS_P1  (WS_CNT + N_NODES)                 // 650000 (8B aligned: *4 % 8 == 0)
#define WS_P2  (WS_P1 + 8 * 32 * 32 * 2)          // 666384
// total floats: WS_P2 + 64*32*32*2 = 797456  (~3.2 MB)

__device__ __forceinline__ v8f wmma_f32_k4(v2f a, v2f b, v8f c) {
  // D = A(16x4,f32) * B(4x16,f32) + C(16x16,f32)
  return __builtin_amdgcn_wmma_f32_16x16x4_f32(false, a, false, b, (short)0, c,
                                               false, false);
}

__global__ void __launch_bounds__(256)
tp_zero_ws(float* __restrict__ ws, int n) {
  int i = blockIdx.x * blockDim.x + threadIdx.x;
  if (i < n) ws[i] = 0.0f;
}

// Repack W1 (128x128) into WMMA B-fragment order:
// p1[(nb*32 + k4)*32 + lane] = float2{ W1[(4k4+2kh)*128 + nb*16+col],
//                                      W1[(4k4+2kh+1)*128 + nb*16+col] }
__global__ void __launch_bounds__(256)
tp_pack_w1(const float* __restrict__ w1, float* __restrict__ p1) {
  int idx = blockIdx.x * blockDim.x + threadIdx.x;   // 8192 threads
  if (idx >= 8 * 32 * 32) return;
  const int lane = idx & 31;
  const int k4   = (idx >> 5) & 31;
  const int nb   = idx >> 10;
  const int col  = lane & 15;
  const int kh   = lane >> 4;
  const int krow = 4 * k4 + 2 * kh;
  p1[idx * 2 + 0] = w1[(size_t)krow * 128 + nb * 16 + col];
  p1[idx * 2 + 1] = w1[(size_t)(krow + 1) * 128 + nb * 16 + col];
}

// Repack W2 (128x1024) the same way, 64 column chunks of 16.
__global__ void __launch_bounds__(256)
tp_pack_w2(const float* __restrict__ w2, float* __restrict__ p2) {
  int idx = blockIdx.x * blockDim.x + threadIdx.x;   // 65536 threads
  if (idx >= 64 * 32 * 32) return;
  const int lane = idx & 31;
  const int k4   = (idx >> 5) & 31;
  const int ci   = idx >> 10;
  const int col  = lane & 15;
  const int kh   = lane >> 4;
  const int krow = 4 * k4 + 2 * kh;
  p2[idx * 2 + 0] = w2[(size_t)krow * 1024 + ci * 16 + col];
  p2[idx * 2 + 1] = w2[(size_t)(krow + 1) * 1024 + ci * 16 + col];
}

__global__ void __launch_bounds__(256)
tp_agg_main(const float* __restrict__ agg_node_attr,  // [E,64]
            const int*   __restrict__ agg_index,      // [E]
            const float* __restrict__ edge_attr,      // [E,128]
            const float* __restrict__ edge_sh,        // [E,4]
            const float* __restrict__ fc_b1,          // [128]
            const float* __restrict__ fc_b2,          // [1024]
            const float* __restrict__ w1p,            // packed W1 frags
            const float* __restrict__ w2p,            // packed W2 frags
            float* __restrict__ node_sum,             // [N_NODES*64]
            float* __restrict__ node_cnt)             // [N_NODES]
{
  __shared__ float sA[TE * LDA];  // edge_attr tile; reused for tp coefficients
  __shared__ float sH[TE * LDA];  // hidden tile h = relu(A@W1+b1)
  __shared__ float sP[TE * 64];   // partial tp accumulators from cg==1 waves

  const int tid  = threadIdx.x;
  const int lane = tid & 31;
  const int wv   = tid >> 5;   // wave 0..7
  const int col  = lane & 15;  // fragment M/N index
  const int kh   = lane >> 4;  // K-half (0/1)
  const int eBase = blockIdx.x * TE;

  // ---- stage edge_attr tile (64 x 128 f32) into LDS via Tensor Data Mover ----
  // D# describes a 64x128 f32 2D tile, row stride 128 elements, with LDS
  // padding of 4 DWORDs every 128 DWORDs => LDS row stride 132 floats == LDA.
#if defined(__has_builtin) && __has_builtin(__builtin_amdgcn_tensor_load_to_lds)
  if (wv == 0) {
    const unsigned lds_base = (unsigned)(uintptr_t)(void*)sA;  // LDS byte offset
    const unsigned long long ga =
        (unsigned long long)(uintptr_t)(edge_attr + (size_t)eBase * 128);
    u32x4 g0;
    g0[0] = 1u;                                   // count=1, user mode
    g0[1] = lds_base;                             // lds_addr
    g0[2] = (unsigned)(ga & 0xFFFFFFFFu);         // global_addr[31:0]
    g0[3] = (unsigned)((ga >> 32) & 0x01FFFFFFu)  // global_addr[56:32]
            | (2u << 30);                         // type = 2 (image)
    i32x8 g1;
    g1[0] = (2 << 16)      // data_size = 4 bytes
          | (1 << 20)      // pad_enable
          | (6 << 22)      // pad_interval: 128 DWORDs
          | (3 << 25);     // pad_amount:   4 DWORDs
    g1[1] = (128 << 16);   // tensor_dim0 = 128
    g1[2] = (64 << 16);    // tensor_dim1 = 64
    g1[3] = (128 << 16);   // tile_dim0 = 128
    g1[4] = 64;            // tile_dim1 = 64, tile_dim2 = 0
    g1[5] = 128;           // tensor_dim0_stride = 128 elements
    g1[6] = 0;
    g1[7] = 0;
    i32x4 gz4 = {0, 0, 0, 0};
    i32x8 gz8 = {0, 0, 0, 0, 0, 0, 0, 0};
    __builtin_amdgcn_tensor_load_to_lds(g0, g1, gz4, gz4, gz8, 0);
    __builtin_amdgcn_s_wait_tensorcnt(0);
  }
#else
  for (int i = 0; i < 8; ++i) {
    int idx = tid + i * 256;
    int row = idx >> 5;
    int c4  = (idx & 31) << 2;
    const float4 v =
        *(const float4*)(edge_attr + (size_t)(eBase + row) * 128 + c4);
    sA[row * LDA + c4 + 0] = v.x;
    sA[row * LDA + c4 + 1] = v.y;
    sA[row * LDA + c4 + 2] = v.z;
    sA[row * LDA + c4 + 3] = v.w;
  }
#endif
  __syncthreads();

  // ---- GEMM1: h = relu(A @ W1 + b1); wave wv owns h columns [16wv,16wv+16) ----
  {
    const int n0 = 16 * wv;
    const float bias = fc_b1[n0 + col];
    v8f c1[4];
    for (int mt = 0; mt < 4; ++mt)
      for (int r = 0; r < 8; ++r) c1[mt][r] = bias;

    const v2f* bfr = (const v2f*)w1p + (size_t)wv * 32 * 32 + lane;
    for (int k4 = 0; k4 < 32; ++k4) {
      const int krow = 4 * k4 + 2 * kh;
      const v2f b = bfr[k4 * 32];               // coalesced global_load_b64
      for (int mt = 0; mt < 4; ++mt) {
        v2f a;
        a.x = sA[(mt * 16 + col) * LDA + krow];
        a.y = sA[(mt * 16 + col) * LDA + krow + 1];
        c1[mt] = wmma_f32_k4(a, b, c1[mt]);
      }
    }
    for (int mt = 0; mt < 4; ++mt)
      for (int r = 0; r < 8; ++r) {
        const int row = mt * 16 + r + 8 * kh;
        const float v = c1[mt][r];
        sH[row * LDA + n0 + col] = v > 0.0f ? v : 0.0f;
      }
  }
  __syncthreads();

  // ---- per-edge tensor-product coefficients (constants baked in) -> sA ----
  // layout per edge (128 floats): [0:16)=c000(u)  [16:32)=c110(u)
  //                               [32:80)=c011(u,m) [80:128)=c101(u,m)
  {
    const int e_loc = tid >> 2;  // 4 threads per edge
    const int part  = tid & 3;
    const int e = eBase + e_loc;
    const float s0  = edge_sh[e * 4 + 0];
    const float s1a = edge_sh[e * 4 + 1];
    const float s1b = edge_sh[e * 4 + 2];
    const float s1c = edge_sh[e * 4 + 3];
    float* cf = sA + e_loc * LDA;
    for (int uu = 0; uu < 4; ++uu) {
      const int u = part * 4 + uu;
      const float x0  = agg_node_attr[(size_t)e * 64 + u];
      const float x1a = agg_node_attr[(size_t)e * 64 + 16 + u * 3 + 0];
      const float x1b = agg_node_attr[(size_t)e * 64 + 16 + u * 3 + 1];
      const float x1c = agg_node_attr[(size_t)e * 64 + 16 + u * 3 + 2];
      cf[u]      = PW0 * s0 * x0;
      cf[16 + u] = PW0I3 * (x1a * s1a + x1b * s1b + x1c * s1c);
      cf[32 + u * 3 + 0] = PW1I3 * x0 * s1a;
      cf[32 + u * 3 + 1] = PW1I3 * x0 * s1b;
      cf[32 + u * 3 + 2] = PW1I3 * x0 * s1c;
      cf[80 + u * 3 + 0] = PW1I3 * s0 * x1a;
      cf[80 + u * 3 + 1] = PW1I3 * s0 * x1b;
      cf[80 + u * 3 + 2] = PW1I3 * s0 * x1c;
    }
  }
  __syncthreads();

  // ---- GEMM2 fused with tensor-product contraction ----
  // wave wv: edge subtile mt = wv&3, chunk parity cg = wv>>2 (32 chunks each)
  const int mt = wv & 3;
  const int cg = wv >> 2;
  v8f t0, t1a, t1b, t1c;
  for (int r = 0; r < 8; ++r) { t0[r] = 0.f; t1a[r] = 0.f; t1b[r] = 0.f; t1c[r] = 0.f; }

  const float* hrow = sH + (mt * 16 + col) * LDA;  // this lane's A-frag row
  const v2f* w2frag = (const v2f*)w2p;
  for (int ci = cg; ci < 64; ci += 2) {
    const int p  = ci >> 4;    // path 0..3
    const int u  = ci & 15;    // u index
    const v2f* bfr = w2frag + (size_t)ci * 32 * 32 + lane;
    if (ci + 2 < 64)           // prefetch next chunk's packed fragments
      __builtin_prefetch(w2frag + (size_t)(ci + 2) * 32 * 32 + lane, 0, 1);

    v8f c2;
    const float bias = fc_b2[ci * 16 + col];
    for (int r = 0; r < 8; ++r) c2[r] = bias;    // exact b2 handling

    for (int k4 = 0; k4 < 32; ++k4) {
      const int krow = 4 * k4 + 2 * kh;
      v2f a;
      a.x = hrow[krow];
      a.y = hrow[krow + 1];
      const v2f b = bfr[k4 * 32];                // coalesced global_load_b64
      c2 = wmma_f32_k4(a, b, c2);
    }

    // contract this (p,u) slice against per-edge coefficients
    const int erb = (mt * 16 + 8 * kh) * LDA;    // coef row base at r=0
    if (p == 0) {
      for (int r = 0; r < 8; ++r) t0[r] += sA[erb + r * LDA + u] * c2[r];
    } else if (p == 3) {
      for (int r = 0; r < 8; ++r) t0[r] += sA[erb + r * LDA + 16 + u] * c2[r];
    } else if (p == 1) {
      for (int r = 0; r < 8; ++r) {
        const float* cf = sA + erb + r * LDA + 32 + u * 3;
        t1a[r] += cf[0] * c2[r];
        t1b[r] += cf[1] * c2[r];
        t1c[r] += cf[2] * c2[r];
      }
    } else {
      for (int r = 0; r < 8; ++r) {
        const float* cf = sA + erb + r * LDA + 80 + u * 3;
        t1a[r] += cf[0] * c2[r];
        t1b[r] += cf[1] * c2[r];
        t1c[r] += cf[2] * c2[r];
      }
    }
  }

  // ---- combine the two partial waves per edge subtile via LDS ----
  if (cg == 1) {
    for (int r = 0; r < 8; ++r) {
      const int row = r + 8 * kh;
      float* pr = sP + (mt * 16 + row) * 64;
      pr[col] = t0[r];
      pr[16 + col * 3 + 0] = t1a[r];
      pr[16 + col * 3 + 1] = t1b[r];
      pr[16 + col * 3 + 2] = t1c[r];
    }
  }
  __syncthreads();
  if (cg == 0) {
    for (int r = 0; r < 8; ++r) {
      const int row   = r + 8 * kh;
      const int e_row = mt * 16 + row;
      const int e     = eBase + e_row;
      const float* pr = sP + e_row * 64;
      const int node  = agg_index[e];
      float* drow = node_sum + (size_t)node * 64;
      atomicAdd(drow + col, t0[r] + pr[col]);
      atomicAdd(drow + 16 + col * 3 + 0, t1a[r] + pr[16 + col * 3 + 0]);
      atomicAdd(drow + 16 + col * 3 + 1, t1b[r] + pr[16 + col * 3 + 1]);
      atomicAdd(drow + 16 + col * 3 + 2, t1c[r] + pr[16 + col * 3 + 2]);
      if (col == 0) atomicAdd(node_cnt + node, 1.0f);
    }
  }
}

__global__ void __launch_bounds__(256)
tp_finalize(const float* __restrict__ node_sum,
            const float* __restrict__ node_cnt,
            const float* __restrict__ dst_node_attr,
            float* __restrict__ out) {
  int i = blockIdx.x * blockDim.x + threadIdx.x;
  if (i < N_NODES * 64) {
    const int node = i >> 6;
    const float c = node_cnt[node];
    const float denom = c > 1.0f ? c : 1.0f;
    out[i] = node_sum[i] / denom + dst_node_attr[i];
  }
}

extern "C" void kernel_launch(void* const* d_in, const int* in_sizes, int n_in,
                              void* d_out, int out_size, void* d_ws, size_t ws_size,
                              hipStream_t stream) {
  const float* dst_node_attr = (const float*)d_in[0];
  const float* agg_node_attr = (const float*)d_in[1];
  const int*   agg_index     = (const int*)d_in[2];
  const float* edge_attr     = (const float*)d_in[3];
  const float* edge_sh       = (const float*)d_in[4];
  const float* fc_w1         = (const float*)d_in[5];
  const float* fc_b1         = (const float*)d_in[6];
  const float* fc_w2         = (const float*)d_in[7];
  const float* fc_b2         = (const float*)d_in[8];
  float* out = (float*)d_out;

  float* ws = (float*)d_ws;
  float* ws_sum = ws + WS_SUM;
  float* ws_cnt = ws + WS_CNT;
  float* ws_p1  = ws + WS_P1;
  float* ws_p2  = ws + WS_P2;

  const int nzero = N_NODES * 64 + N_NODES;
  tp_zero_ws<<<(nzero + 255) / 256, 256, 0, stream>>>(ws, nzero);
  tp_pack_w1<<<(8 * 32 * 32) / 256, 256, 0, stream>>>(fc_w1, ws_p1);
  tp_pack_w2<<<(64 * 32 * 32) / 256, 256, 0, stream>>>(fc_w2, ws_p2);

  tp_agg_main<<<N_EDGES / TE, 256, 0, stream>>>(
      agg_node_attr, agg_index, edge_attr, edge_sh,
      fc_b1, fc_b2, ws_p1, ws_p2, ws_sum, ws_cnt);

  tp_finalize<<<(N_NODES * 64 + 255) / 256, 256, 0, stream>>>(
      ws_sum, ws_cnt, dst_node_attr, out);
}